// CostYoloV2_59416577573083
// MI455X (gfx1250) — compile-verified
//
#include <hip/hip_runtime.h>
#include <hip/hip_bf16.h>

// YOLOv2 cost on MI455X (gfx1250, wave32).
// Observations driving the design:
//  - loss_noobj / loss_coords only use batch B-1 => IoU grid only for b=63.
//  - Total live traffic ~ a few MB, FLOPs ~ 5M: latency/launch bound.
// CDNA5 features used:
//  - TDM tensor_load_to_lds (inline asm descriptor) + s_wait_tensorcnt to
//    stage the batch-63 truth tile into LDS.
//  - V_WMMA_F32_16X16X4_F32 (ones-matrix trick) for deterministic 64-way
//    reductions (block partials and final), EXEC all-ones guaranteed.
//  - global_prefetch_b8 via __builtin_prefetch on gather rows.

#define NUMA   5
#define CLASSES 20
#define ALc    25          // CLASSES + 4 + 1
#define Bc     64
#define Hc     32
#define Wc     32
#define HWc    1024
#define Tc     50

typedef __attribute__((ext_vector_type(2))) float        v2f;
typedef __attribute__((ext_vector_type(8))) float        v8f;
typedef __attribute__((ext_vector_type(4))) unsigned int v4u;
typedef __attribute__((ext_vector_type(8))) unsigned int v8u;

// ---------------------------------------------------------------------------
// Sum src[0..63] across one full wave32 using a single WMMA:
//   D = A(16x4 = data) x B(4x16 = ones)  =>  D[m,n] = rowsum_m.
// A layout (32-bit A 16x4): lane L<16 supplies A[L,0],A[L,1]; lane L+16
// supplies A[L,2],A[L,3]  => A[m,k] = src[4m+k].
// C/D layout: lane 0 holds D[0..7, 0] in its 8 dest VGPRs, lane 16 holds
// D[8..15, 0]. Must be called with all 32 lanes of the wave active.
// ---------------------------------------------------------------------------
__device__ __forceinline__ float wave_reduce64_wmma(const float* __restrict__ src,
                                                    int lane) {
#if __has_builtin(__builtin_amdgcn_wmma_f32_16x16x4_f32)
  v2f a, bo;
  if (lane < 16) { a.x = src[4 * lane + 0];        a.y = src[4 * lane + 1]; }
  else           { a.x = src[4 * (lane - 16) + 2]; a.y = src[4 * (lane - 16) + 3]; }
  bo.x = 1.0f; bo.y = 1.0f;
  v8f c = {0.f, 0.f, 0.f, 0.f, 0.f, 0.f, 0.f, 0.f};
  v8f d = __builtin_amdgcn_wmma_f32_16x16x4_f32(
      /*neg_a=*/false, a, /*neg_b=*/false, bo,
      /*c_mod=*/(short)0, c, /*reuse_a=*/false, /*reuse_b=*/false);
  float s = d[0] + d[1] + d[2] + d[3] + d[4] + d[5] + d[6] + d[7];
  return __shfl(s, 0, 32) + __shfl(s, 16, 32);   // D[0..7,0] + D[8..15,0]
#else
  float s = 0.f;
  for (int k = 0; k < 64; ++k) s += src[k];
  return s;
#endif
}

// Deterministic 256 -> 1 block reduction; result valid on thread 0.
__device__ __forceinline__ float block_reduce_256(float v) {
  __shared__ float red[256];
  int tid = threadIdx.x;
  red[tid] = v;
  __syncthreads();
  if (tid < 64) red[tid] += red[tid + 64] + red[tid + 128] + red[tid + 192];
  __syncthreads();
  float out = 0.f;
  if (tid < 32) out = wave_reduce64_wmma(red, tid);  // full wave 0, EXEC=~0
  return out;
}

// ---------------------------------------------------------------------------
// Kernel A: batch-63 noobj loss + coord-warm loss.
// Grid: 20 blocks x 256 threads = 5120 = NUM * HW cells.
// Truth tile (50x5 f32 = 1000B) staged to LDS via Tensor Data Mover.
// ---------------------------------------------------------------------------
__global__ __launch_bounds__(256) void yolo_noobj_kernel(
    const float* __restrict__ x, const float* __restrict__ truth,
    float* __restrict__ ws) {
  __shared__ float tb[Tc * 5];
  const float* tr = truth + (size_t)(Bc - 1) * Tc * 5;

  if (threadIdx.x < 32) {  // wave 0 issues one TDM op for the block
    unsigned long long ga   = (unsigned long long)tr;          // global VA
    unsigned           ldsa = (unsigned)(unsigned long long)(&tb[0]); // LDS byte addr = low 32 bits
    // D# group 0: count=1, lds_addr, global_addr[56:0], type=2 (bits 127:126)
    v4u g0;
    g0[0] = 1u;
    g0[1] = ldsa;
    g0[2] = (unsigned)(ga & 0xffffffffull);
    g0[3] = ((unsigned)((ga >> 32) & 0x1ffffffull)) | 0x80000000u;
    // D# group 1: data_size=4B; tensor_dim0=250, tensor_dim1=1,
    // tile_dim0=250, tile_dim1=1, stride0=250 (1-D contiguous copy).
    v8u g1;
    g1[0] = 2u << 16;                 // data_size = 2 (4 bytes)
    g1[1] = (250u & 0xffffu) << 16;   // tensor_dim0[15:0]
    g1[2] = (250u >> 16) | (1u << 16);// tensor_dim0[31:16]=0, tensor_dim1 lo=1
    g1[3] = (250u << 16);             // tensor_dim1 hi=0, tile_dim0=250
    g1[4] = 1u;                       // tile_dim1=1, tile_dim2=0
    g1[5] = 250u;                     // tensor_dim0_stride[31:0]
    g1[6] = 0u;
    g1[7] = 0u;
    asm volatile("tensor_load_to_lds %0, %1" :: "s"(g0), "s"(g1) : "memory");
    __builtin_amdgcn_s_wait_tensorcnt(0);
  }
  __syncthreads();

  int idx = blockIdx.x * 256 + threadIdx.x;   // 0..5119
  int a   = idx >> 10;                        // anchor
  int hw  = idx & 1023;

  const float* base =
      x + (size_t)(Bc - 1) * (NUMA * ALc) * HWc + (size_t)(a * ALc) * HWc + hw;
  float px = base[0];
  float py = base[HWc];
  float pw = base[2 * HWc];
  float ph = base[3 * HWc];
  float obj = x[(size_t)(Bc - 1) * (NUMA * ALc) * HWc + (size_t)4 * HWc + hw];

  float al_ = px - 0.5f * pw, au_ = py - 0.5f * ph;
  float ar_ = px + 0.5f * pw, ad_ = py + 0.5f * ph;
  float pa  = pw * ph;

  float m = -3.0e38f;
  for (int t = 0; t < Tc; ++t) {               // LDS broadcasts, no conflicts
    float bx = tb[t * 5 + 0], by = tb[t * 5 + 1];
    float bw = tb[t * 5 + 2], bh = tb[t * 5 + 3];
    float bl = bx - 0.5f * bw, bu = by - 0.5f * bh;
    float br = bx + 0.5f * bw, bd = by + 0.5f * bh;
    float iw = fmaxf(fminf(ar_, br) - fmaxf(al_, bl), 0.f);
    float ih = fmaxf(fminf(ad_, bd) - fmaxf(au_, bu), 0.f);
    float inter = iw * ih;
    float uni = pa + bw * bh - inter;
    m = fmaxf(m, inter / uni);
  }

  float acc = (m > 0.5f) ? 0.f : obj * obj;    // obj^2 * (1 - mask)
  float dx = px - 0.5f, dy = py - 0.5f;        // coord-warm term
  acc += 0.01f * (dx * dx + dy * dy + pw * pw + ph * ph);

  float tot = block_reduce_256(acc);
  if (threadIdx.x == 0) ws[blockIdx.x] = tot;  // slots 0..19
}

// ---------------------------------------------------------------------------
// Kernel B: per-(b,t) gathered-cell losses. 64*50 = 3200 items, 13 blocks.
// ---------------------------------------------------------------------------
__global__ __launch_bounds__(256) void yolo_cells_kernel(
    const float* __restrict__ x, const float* __restrict__ truth,
    const float* __restrict__ anchors, float* __restrict__ ws) {
  int idx = blockIdx.x * 256 + threadIdx.x;
  float acc = 0.f;
  if (idx < Bc * Tc) {
    int b = idx / Tc;
    int t = idx - b * Tc;
    const float* tr = truth + (size_t)b * Tc * 5;

    // valid = cumprod over s<=t of (w_t >= 1e-5)
    float valid = 1.f;
    for (int s = 0; s <= t; ++s) valid = (tr[s * 5 + 2] >= 1e-5f) ? valid : 0.f;

    float txt = tr[t * 5 + 0], tyt = tr[t * 5 + 1];
    float wt  = tr[t * 5 + 2], ht  = tr[t * 5 + 3];
    float clsf = tr[t * 5 + 4];

    int ci = (int)(txt * (float)Wc); ci = ci < 0 ? 0 : (ci > Wc - 1 ? Wc - 1 : ci);
    int cj = (int)(tyt * (float)Hc); cj = cj < 0 ? 0 : (cj > Hc - 1 ? Hc - 1 : cj);

    const float* base = x + (size_t)b * (NUMA * ALc) * HWc + (size_t)(cj * Wc + ci);
#pragma unroll
    for (int a2 = 0; a2 < NUMA; ++a2)          // warm gather lines in GL2/L0
      __builtin_prefetch(base + (size_t)(a2 * ALc) * HWc, 0, 1);

    float wp[NUMA], hp[NUMA];
#pragma unroll
    for (int a2 = 0; a2 < NUMA; ++a2) {
      wp[a2] = base[(size_t)(a2 * ALc + 2) * HWc];
      hp[a2] = base[(size_t)(a2 * ALc + 3) * HWc];
    }
    float obj0 = base[(size_t)4 * HWc];        // anchor 0, slot 4

    auto ratio = [&](int a2) {
      float iw = fmaxf(fminf(wt, wp[a2]), 0.f);
      float ih = fmaxf(fminf(ht, hp[a2]), 0.f);
      float inter = iw * ih;
      return inter / (wt * ht + wp[a2] * hp[a2] - inter);
    };
    float bestr = ratio(0);
    int best = 0;
    float wbst = wp[0], hbst = hp[0];
#pragma unroll
    for (int a2 = 1; a2 < NUMA; ++a2) {        // first-max wins (argmax)
      float r = ratio(a2);
      if (r > bestr) { bestr = r; best = a2; wbst = wp[a2]; hbst = hp[a2]; }
    }

    // loss_obj (all batches)
    float e = 1.f - obj0;
    acc += e * e * valid;

    // loss_classes (all batches)
    int ct = (int)clsf; ct = ct < 0 ? 0 : (ct > CLASSES - 1 ? CLASSES - 1 : ct);
    const float* cbase = base + (size_t)(best * ALc + 5) * HWc;
    float cl = 0.f;
#pragma unroll 4
    for (int c = 0; c < CLASSES; ++c) {
      float diff = ((c == ct) ? 1.f : 0.f) - cbase[(size_t)c * HWc];
      cl += diff * diff;
    }
    acc += cl * valid;

    // loss_coords (batch B-1 only)
    if (b == Bc - 1) {
      float txc = (float)ci * (1.f / (float)Wc);
      float tyc = (float)cj * (1.f / (float)Hc);
      float tw  = expf(wt) * anchors[2 * (NUMA - 1) + 0] * (1.f / (float)Wc);
      float th2 = expf(ht) * anchors[2 * (NUMA - 1) + 1] * (1.f / (float)Hc);
      float sc = 2.f - wt * ht;
      float d0 = txc * sc, d1 = tyc * sc;
      float d2 = (tw - wbst) * sc, d3 = (th2 - hbst) * sc;
      acc += (d0 * d0 + d1 * d1 + d2 * d2 + d3 * d3) * valid;
    }
  }
  float tot = block_reduce_256(acc);
  if (threadIdx.x == 0) ws[32 + blockIdx.x] = tot;  // slots 32..44
}

// ---------------------------------------------------------------------------
// Final: one full wave reduces 64 partial slots with one WMMA, writes scalar.
// ---------------------------------------------------------------------------
__global__ __launch_bounds__(32) void yolo_final_reduce(
    const float* __restrict__ ws, float* __restrict__ out) {
  float s = wave_reduce64_wmma(ws, threadIdx.x);
  if (threadIdx.x == 0) out[0] = s;
}

__global__ void yolo_init_ws(float* __restrict__ ws) { ws[threadIdx.x] = 0.f; }

extern "C" void kernel_launch(void* const* d_in, const int* in_sizes, int n_in,
                              void* d_out, int out_size, void* d_ws, size_t ws_size,
                              hipStream_t stream) {
  (void)in_sizes; (void)n_in; (void)out_size; (void)ws_size;
  const float* x       = (const float*)d_in[0];
  const float* truth   = (const float*)d_in[1];
  const float* anchors = (const float*)d_in[2];
  float* out = (float*)d_out;
  float* ws  = (float*)d_ws;   // 64 f32 partial-sum slots

  yolo_init_ws<<<1, 64, 0, stream>>>(ws);
  yolo_noobj_kernel<<<(NUMA * HWc) / 256, 256, 0, stream>>>(x, truth, ws);
  yolo_cells_kernel<<<(Bc * Tc + 255) / 256, 256, 0, stream>>>(x, truth, anchors, ws);
  yolo_final_reduce<<<1, 32, 0, stream>>>(ws, out);
}